// GCNModel2_9363028705695
// MI455X (gfx1250) — compile-verified
//
#include <hip/hip_runtime.h>
#include <hip/hip_bf16.h>

#define NNODES 100000
#define EDGES  1600000
#define DH     128
#define NCLS   100

typedef float v2f __attribute__((ext_vector_type(2)));
typedef float v8f __attribute__((ext_vector_type(8)));

// ---------------------------------------------------------------- degree
__global__ void deg_kernel(const int* __restrict__ dst, float* __restrict__ deg, int E) {
    int i = blockIdx.x * blockDim.x + threadIdx.x;
    if (i < E) atomicAdd(&deg[dst[i]], 1.0f);
}

__global__ void dinv_kernel(const float* __restrict__ deg, float* __restrict__ dinv, int n) {
    int i = blockIdx.x * blockDim.x + threadIdx.x;
    if (i < n) dinv[i] = rsqrtf(deg[i] + 1.0f);
}

// ---------------------------------------------------------------- GEMM via V_WMMA_F32_16X16X4_F32
// out[M x F] = A[M x K] @ W[K x F] (+ bias). Row-major. M % 16 == 0, K % 4 == 0 (compile-time).
// Block = 256 threads (8 waves). Block owns 16 rows; wave w owns cols [16w, 16w+16).
// A tile (16 rows = contiguous 16*K floats) staged in LDS, shared by all waves.
// ISA layouts (CDNA5 ISA):
//   A 16x4 f32 : lanes 0-15 hold row M=lane, {K+0,K+1}; lanes 16-31 hold {K+2,K+3}
//   B  4x16 f32: lanes 0-15 hold col N=lane, {K+0,K+1}; lanes 16-31 hold {K+2,K+3}
//   C/D 16x16  : VGPR r -> M=r (lanes 0-15) / M=r+8 (lanes 16-31), N = lane&15
// For F % 16 != 0, out-of-range lanes CLAMP their W load address (output columns are
// independent in D = A@B + C, so their finite garbage never touches valid columns and
// is simply not stored).
template <int K, int F>
__global__ void gemm_wmma_f32(const float* __restrict__ A, const float* __restrict__ W,
                              const float* __restrict__ bias, float* __restrict__ out) {
    __shared__ float sA[16 * K];
    const int rowBase = blockIdx.x * 16;

    // contiguous coalesced stage of the 16 x K A-tile
    {
        const float4* __restrict__ g4 = (const float4*)(A + (size_t)rowBase * K);
        float4* __restrict__ s4 = (float4*)sA;
        constexpr int TOT4 = (16 * K) / 4;
        #pragma unroll
        for (int i = 0; i < (TOT4 + 255) / 256; ++i) {
            int idx = threadIdx.x + i * 256;
            if (TOT4 % 256 == 0 || idx < TOT4) s4[idx] = g4[idx];
        }
    }
    __syncthreads();

    constexpr int NT = (F + 15) / 16;     // number of 16-wide column tiles
    const int wave = threadIdx.x >> 5;
    if (wave >= NT) return;               // whole-wave exit; EXEC all-1s for WMMA waves

    const int lane = threadIdx.x & 31;
    const int hi   = lane >> 4;           // 0: K+{0,1}, 1: K+{2,3}
    const int l15  = lane & 15;
    const int col  = wave * 16 + l15;
    const int colc = (F % 16 == 0) ? col : (col < F ? col : F - 1);  // clamp, no branch in loop

    const float* __restrict__ sArow = sA + l15 * K;
    const float* __restrict__ Wcol  = W + colc;

    v8f acc = {};
    #pragma unroll
    for (int k = 0; k < K; k += 4) {
        const int kk = k + 2 * hi;
        v2f a, b;
        a.x = sArow[kk];
        a.y = sArow[kk + 1];
        b.x = Wcol[(size_t)kk * F];
        b.y = Wcol[(size_t)(kk + 1) * F];
        acc = __builtin_amdgcn_wmma_f32_16x16x4_f32(
            /*neg_a=*/false, a, /*neg_b=*/false, b,
            /*c_mod=*/(short)0, acc, /*reuse_a=*/false, /*reuse_b=*/false);
    }

    if (F % 16 == 0 || col < F) {
        const float bv  = bias ? bias[col] : 0.0f;
        const int   row = rowBase + 8 * hi;
        #pragma unroll
        for (int r = 0; r < 8; ++r)
            out[(size_t)(row + r) * F + col] = acc[r] + bv;
    }
}

// ---------------------------------------------------------------- per-edge scatter (one wave / edge)
__global__ void edge_agg(const float* __restrict__ h, const int* __restrict__ src,
                         const int* __restrict__ dst, const float* __restrict__ dinv,
                         float* __restrict__ agg, int E, int F) {
    const int e    = (blockIdx.x * blockDim.x + threadIdx.x) >> 5;
    const int lane = threadIdx.x & 31;
    if (e >= E) return;
    const int   s    = src[e];
    const int   d    = dst[e];
    const float coef = dinv[s] * dinv[d];
    const int   f    = lane * 4;
    if (f < F) {
        const float4 v = *(const float4*)(h + (size_t)s * F + f);
        float* a = agg + (size_t)d * F + f;
        atomicAdd(a + 0, v.x * coef);
        atomicAdd(a + 1, v.y * coef);
        atomicAdd(a + 2, v.z * coef);
        atomicAdd(a + 3, v.w * coef);
    }
}

// ---------------------------------------------------------------- self-loop + bias (+relu)
__global__ void finalize_kernel(const float* __restrict__ agg, const float* __restrict__ h,
                                const float* __restrict__ dinv, const float* __restrict__ b,
                                float* __restrict__ out, int n, int F, int do_relu) {
    const long long i   = (long long)blockIdx.x * blockDim.x + threadIdx.x;
    const long long tot = (long long)n * F;
    if (i >= tot) return;
    const int row = (int)(i / F);
    const int col = (int)(i - (long long)row * F);
    const float di = dinv[row];
    float v = agg[i] + h[i] * di * di + b[col];
    out[i] = do_relu ? fmaxf(v, 0.0f) : v;
}

// ---------------------------------------------------------------- host side
static void run_agg_epilogue(const float* bufH, float* bufAgg, float* outBuf,
                             const int* src, const int* dst, const float* dinv,
                             const float* b, int F, int relu, hipStream_t stream) {
    hipMemsetAsync(bufAgg, 0, (size_t)NNODES * F * sizeof(float), stream);
    const int eblocks = (EDGES * 32 + 255) / 256;
    edge_agg<<<eblocks, 256, 0, stream>>>(bufH, src, dst, dinv, bufAgg, EDGES, F);
    const long long tot = (long long)NNODES * F;
    finalize_kernel<<<(int)((tot + 255) / 256), 256, 0, stream>>>(bufAgg, bufH, dinv, b, outBuf,
                                                                  NNODES, F, relu);
}

extern "C" void kernel_launch(void* const* d_in, const int* in_sizes, int n_in,
                              void* d_out, int out_size, void* d_ws, size_t ws_size,
                              hipStream_t stream) {
    const float* x  = (const float*)d_in[0];
    const int*   ei = (const int*)d_in[1];
    const float* W1 = (const float*)d_in[2];   const float* b1 = (const float*)d_in[3];
    const float* W2 = (const float*)d_in[4];   const float* b2 = (const float*)d_in[5];
    const float* W3 = (const float*)d_in[6];   const float* b3 = (const float*)d_in[7];
    const float* W4 = (const float*)d_in[8];   const float* b4 = (const float*)d_in[9];
    const float* W5 = (const float*)d_in[10];  const float* b5 = (const float*)d_in[11];
    const float* Wl = (const float*)d_in[12];  const float* bl = (const float*)d_in[13];

    const int* src = ei;
    const int* dst = ei + EDGES;

    float* ws     = (float*)d_ws;
    float* deg    = ws;                                    // N
    float* dinv   = ws + NNODES;                           // N
    float* bufH   = ws + 2 * (size_t)NNODES;               // N*128
    float* bufAgg = bufH + (size_t)NNODES * DH;            // N*128
    float* bufX   = bufAgg + (size_t)NNODES * DH;          // N*128

    // degrees (incl. self loop) -> dinv
    hipMemsetAsync(deg, 0, NNODES * sizeof(float), stream);
    deg_kernel<<<(EDGES + 255) / 256, 256, 0, stream>>>(dst, deg, EDGES);
    dinv_kernel<<<(NNODES + 255) / 256, 256, 0, stream>>>(deg, dinv, NNODES);

    const int gblocks = NNODES / 16;

    // layer 1 (relu)
    gemm_wmma_f32<DH, DH><<<gblocks, 256, 0, stream>>>(x, W1, nullptr, bufH);
    run_agg_epilogue(bufH, bufAgg, bufX, src, dst, dinv, b1, DH, 1, stream);
    // layers 2-4
    gemm_wmma_f32<DH, DH><<<gblocks, 256, 0, stream>>>(bufX, W2, nullptr, bufH);
    run_agg_epilogue(bufH, bufAgg, bufX, src, dst, dinv, b2, DH, 0, stream);
    gemm_wmma_f32<DH, DH><<<gblocks, 256, 0, stream>>>(bufX, W3, nullptr, bufH);
    run_agg_epilogue(bufH, bufAgg, bufX, src, dst, dinv, b3, DH, 0, stream);
    gemm_wmma_f32<DH, DH><<<gblocks, 256, 0, stream>>>(bufX, W4, nullptr, bufH);
    run_agg_epilogue(bufH, bufAgg, bufX, src, dst, dinv, b4, DH, 0, stream);
    // layer 5: 128 -> 100
    gemm_wmma_f32<DH, NCLS><<<gblocks, 256, 0, stream>>>(bufX, W5, nullptr, bufH);
    run_agg_epilogue(bufH, bufAgg, bufX, src, dst, dinv, b5, NCLS, 0, stream);

    // final linear: out = h5 @ Wl + bl  (bias fused into GEMM epilogue)
    gemm_wmma_f32<NCLS, NCLS><<<gblocks, 256, 0, stream>>>(bufX, Wl, bl, (float*)d_out);
}